// MonoEncoder_48163763257994
// MI455X (gfx1250) — compile-verified
//
#include <hip/hip_runtime.h>

#define D_DIM 512
#define C_DIM 4096
#define DROP_P 0.2f
#define INV_KEEP 1.25f   // 1 / (1 - 0.2)

typedef __attribute__((ext_vector_type(2))) float v2f;
typedef __attribute__((ext_vector_type(8))) float v8f;

// ---------------------------------------------------------------------------
// Kernel 1: w[j] = sum_k co[k,j] * cumprod_{k'<=k} mask[k',j]
// mask[k,j] = (mask_u[k,j] > 0.2) ? 1.25 : 0
// One thread per column j; loads of co/mask are coalesced across j.
// Exact early exit: once cum==0 in every lane of the wave, all remaining
// contributions are exactly zero (cumprod is absorbing at 0).
// Also zero-initializes the dim_emb region of d_out (harness poisons it).
// ---------------------------------------------------------------------------
__global__ void __launch_bounds__(128)
compute_w_kernel(const float* __restrict__ co,
                 const float* __restrict__ mask_u,
                 float* __restrict__ w,
                 float* __restrict__ dim_out) {
    const int j = blockIdx.x * blockDim.x + threadIdx.x;
    if (j < D_DIM) dim_out[j] = 0.0f;          // zero dim_emb accumulator
    if (j >= C_DIM) return;

    float cum = 1.0f;
    float acc = 0.0f;
    for (int k = 0; k < C_DIM; ++k) {
        const float mu = mask_u[(size_t)k * C_DIM + j];
        const float m  = (mu > DROP_P) ? INV_KEEP : 0.0f;
        cum *= m;
        acc = fmaf(co[(size_t)k * C_DIM + j], cum, acc);
        if (!__any(cum != 0.0f)) break;        // wave-uniform, numerically exact
    }
    w[j] = acc;
}

// ---------------------------------------------------------------------------
// Kernel 2: coalesced float4 copy of label_emb -> d_out[0 : D*C]
// ---------------------------------------------------------------------------
__global__ void __launch_bounds__(256)
copy_emb_kernel(const float4* __restrict__ src, float4* __restrict__ dst, int n4) {
    const int i = blockIdx.x * blockDim.x + threadIdx.x;
    if (i < n4) dst[i] = src[i];
}

// ---------------------------------------------------------------------------
// Kernel 3: dim_emb = label_emb @ w via V_WMMA_F32_16X16X4_F32 (exact fp32).
//   grid = (D/16 tiles, 8 K-splits), block = 1 wave (32 lanes).
//   A (16x4): A[m][k] = label_emb[tile*16+m][kbase+k]
//     ISA layout: lanes 0-15 -> M=lane, v0=K0, v1=K1; lanes 16-31 -> v0=K2, v1=K3.
//   B (4x16): w broadcast across all 16 N columns -> every D column equals the
//     dot-product tile; B layout mirrors C/D striping (v0: K0|K2, v1: K1|K3).
//   D (16x16): lane0 v[r] = D[r][0] (M=0..7), lane16 v[r] = D[8+r][0].
//   K split across blockIdx.y; partials combined with global f32 atomics.
// ---------------------------------------------------------------------------
__global__ void __launch_bounds__(32)
gemv_wmma_kernel(const float* __restrict__ emb,
                 const float* __restrict__ w,
                 float* __restrict__ dim_out) {
    const int tile   = blockIdx.x;            // 0..31
    const int ksplit = blockIdx.y;            // 0..7
    const int lane   = threadIdx.x;           // 0..31
    const int row    = tile * 16 + (lane & 15);
    const int koff   = (lane < 16) ? 0 : 2;   // half-wave K offset per ISA A layout

    const float* __restrict__ rowp = emb + (size_t)row * C_DIM;
    const int KSPAN = C_DIM / 8;              // 512
    const int k0 = ksplit * KSPAN;

    v8f acc0 = {};
    v8f acc1 = {};
    for (int k = k0; k < k0 + KSPAN; k += 8) {
        v2f a0, b0, a1, b1;
        a0.x = rowp[k + koff];         a0.y = rowp[k + koff + 1];
        b0.x = w[k + koff];            b0.y = w[k + koff + 1];
        a1.x = rowp[k + 4 + koff];     a1.y = rowp[k + 4 + koff + 1];
        b1.x = w[k + 4 + koff];        b1.y = w[k + 4 + koff + 1];
        // 8 args: (neg_a, A, neg_b, B, c_mod, C, reuse_a, reuse_b)
        acc0 = __builtin_amdgcn_wmma_f32_16x16x4_f32(false, a0, false, b0,
                                                     (short)0, acc0, false, false);
        acc1 = __builtin_amdgcn_wmma_f32_16x16x4_f32(false, a1, false, b1,
                                                     (short)0, acc1, false, false);
    }

    float acc[8];
#pragma unroll
    for (int r = 0; r < 8; ++r) acc[r] = acc0[r] + acc1[r];

    if (lane == 0) {
#pragma unroll
        for (int r = 0; r < 8; ++r) atomicAdd(&dim_out[tile * 16 + r], acc[r]);
    } else if (lane == 16) {
#pragma unroll
        for (int r = 0; r < 8; ++r) atomicAdd(&dim_out[tile * 16 + 8 + r], acc[r]);
    }
}

// ---------------------------------------------------------------------------
extern "C" void kernel_launch(void* const* d_in, const int* in_sizes, int n_in,
                              void* d_out, int out_size, void* d_ws, size_t ws_size,
                              hipStream_t stream) {
    const float* label_emb = (const float*)d_in[0];   // (D, C)
    const float* co        = (const float*)d_in[1];   // (C, C)
    const float* mask_u    = (const float*)d_in[2];   // (C, C)

    float* out_emb = (float*)d_out;                          // D*C floats
    float* out_dim = out_emb + (size_t)D_DIM * C_DIM;        // D floats
    float* w       = (float*)d_ws;                           // C floats scratch

    // 1) column scan -> w ; zero dim_emb accumulator
    compute_w_kernel<<<C_DIM / 128, 128, 0, stream>>>(co, mask_u, w, out_dim);

    // 2) copy label_emb into output (coalesced float4)
    const int n4 = (D_DIM * C_DIM) / 4;
    copy_emb_kernel<<<(n4 + 255) / 256, 256, 0, stream>>>(
        (const float4*)label_emb, (float4*)out_emb, n4);

    // 3) GEMV on the WMMA pipe, K split 8 ways + atomic finalize
    dim3 grid(D_DIM / 16, 8);
    gemv_wmma_kernel<<<grid, 32, 0, stream>>>(label_emb, w, out_dim);
}